// KGATLayer_40063454937408
// MI455X (gfx1250) — compile-verified
//
#include <hip/hip_runtime.h>
#include <hip/hip_bf16.h>

#define NN 100000   // nodes
#define NE 500000   // edges
#define NR 200      // relations
#define DD 128      // dim

typedef float v8f __attribute__((ext_vector_type(8)));
typedef float v2f __attribute__((ext_vector_type(2)));

// ---------------- workspace layout (in floats) ----------------
// A_dst   : NN*DD
// A_rel   : NN*DD
// b_sum   : NN
// M1P,M2P : pair-interleaved 128x128 (MP[(k>>1)*256 + j*2 + (k&1)] = M[k][j])
// M3      : row-major 128x128
// bias_c  : DD ; w1,w2 : DD each ; base : 4 ; t3a : NR
// T3, rel_proj, R_out : NR*DD each
#define OFF_ADST   ((size_t)0)
#define OFF_AREL   (OFF_ADST + (size_t)NN*DD)
#define OFF_BSUM   (OFF_AREL + (size_t)NN*DD)
#define OFF_M1P    (OFF_BSUM + NN)
#define OFF_M2P    (OFF_M1P + DD*DD)
#define OFF_M3     (OFF_M2P + DD*DD)
#define OFF_BIASC  (OFF_M3 + DD*DD)
#define OFF_W1     (OFF_BIASC + DD)
#define OFF_W2     (OFF_W1 + DD)
#define OFF_BASE   (OFF_W2 + DD)
#define OFF_T3A    (OFF_BASE + 4)
#define OFF_T3     (OFF_T3A + NR)
#define OFF_RPROJ  (OFF_T3 + (size_t)NR*DD)
#define OFF_ROUT   (OFF_RPROJ + (size_t)NR*DD)

// pair-interleaved index for M[k][j], k,j in [0,128)
__device__ __forceinline__ size_t mp_idx(int k, int j) {
    return (size_t)(k >> 1) * 256 + j * 2 + (k & 1);
}

// ---------------- zero init ----------------
__global__ void kgat_zero(float* __restrict__ p, size_t n) {
    size_t i = (size_t)blockIdx.x * blockDim.x + threadIdx.x;
    size_t stride = (size_t)gridDim.x * blockDim.x;
    for (; i < n; i += stride) p[i] = 0.0f;
}

// ---------------- precompute M1/M2 (packed), M3 (row-major), bias_c ----------------
__global__ void kgat_prep_M(const float* __restrict__ W_ent, const float* __restrict__ W_rel,
                            const float* __restrict__ W_fc,  const float* __restrict__ b_ent,
                            const float* __restrict__ b_rel, const float* __restrict__ b_fc,
                            float* __restrict__ ws) {
    int tid = blockIdx.x * blockDim.x + threadIdx.x;
    if (tid < 3 * DD * DD) {
        int m = tid >> 14;            // which matrix
        int i = (tid >> 7) & 127;     // output row (k index of M)
        int j = tid & 127;            // output col
        const float* A = (m == 2) ? W_rel : W_ent;
        const float* B = W_fc + (size_t)m * DD * DD;   // W_fc rows m*128 .. m*128+127
        float acc = 0.0f;
        for (int k = 0; k < DD; ++k) acc += A[i * DD + k] * B[k * DD + j];
        if (m == 0)      ws[OFF_M1P + mp_idx(i, j)] = acc;
        else if (m == 1) ws[OFF_M2P + mp_idx(i, j)] = acc;
        else             ws[OFF_M3 + i * DD + j] = acc;
    } else if (tid < 3 * DD * DD + DD) {
        int j = tid - 3 * DD * DD;
        float acc = b_fc[j];
        for (int k = 0; k < DD; ++k) {
            acc += b_ent[k] * (W_fc[k * DD + j] + W_fc[(DD + k) * DD + j]);
            acc += b_rel[k] * W_fc[(2 * DD + k) * DD + j];
        }
        ws[OFF_BIASC + j] = acc;
    }
}

// ---------------- precompute w1 = M1@W_a, w2 = M2@W_a, base = bias_c@W_a + b_a ----------------
__global__ void kgat_prep_w(const float* __restrict__ W_a, const float* __restrict__ b_a,
                            float* __restrict__ ws) {
    int tid = blockIdx.x * blockDim.x + threadIdx.x;
    if (tid < DD) {
        float acc = 0.0f;
        for (int j = 0; j < DD; ++j) acc += ws[OFF_M1P + mp_idx(tid, j)] * W_a[j];
        ws[OFF_W1 + tid] = acc;
    } else if (tid < 2 * DD) {
        int i = tid - DD;
        float acc = 0.0f;
        for (int j = 0; j < DD; ++j) acc += ws[OFF_M2P + mp_idx(i, j)] * W_a[j];
        ws[OFF_W2 + i] = acc;
    } else if (tid == 2 * DD) {
        float acc = b_a[0];
        for (int j = 0; j < DD; ++j) acc += ws[OFF_BIASC + j] * W_a[j];
        ws[OFF_BASE] = acc;
    }
}

// ---------------- precompute T3 = rel_embed@M3 and rel_proj = rel_embed@W_rel + b_rel ----------------
__global__ void kgat_prep_relA(const float* __restrict__ rel_embed, const float* __restrict__ W_rel,
                               const float* __restrict__ b_rel, float* __restrict__ ws) {
    int tid = blockIdx.x * blockDim.x + threadIdx.x;
    if (tid < NR * DD) {
        int r = tid >> 7, j = tid & 127;
        float acc = 0.0f;
        for (int k = 0; k < DD; ++k) acc += rel_embed[r * DD + k] * ws[OFF_M3 + k * DD + j];
        ws[OFF_T3 + tid] = acc;
    } else if (tid < 2 * NR * DD) {
        int t = tid - NR * DD;
        int r = t >> 7, j = t & 127;
        float acc = b_rel[j];
        for (int k = 0; k < DD; ++k) acc += rel_embed[r * DD + k] * W_rel[k * DD + j];
        ws[OFF_RPROJ + t] = acc;
    }
}

// ---------------- precompute R_out = rel_proj@W_rel2 + b_rel2 and t3a = T3@W_a + base ----------------
__global__ void kgat_prep_relB(const float* __restrict__ W_rel2, const float* __restrict__ b_rel2,
                               const float* __restrict__ W_a, float* __restrict__ ws) {
    int tid = blockIdx.x * blockDim.x + threadIdx.x;
    if (tid < NR * DD) {
        int r = tid >> 7, j = tid & 127;
        float acc = b_rel2[j];
        for (int k = 0; k < DD; ++k) acc += ws[OFF_RPROJ + r * DD + k] * W_rel2[k * DD + j];
        ws[OFF_ROUT + tid] = acc;
    } else if (tid < NR * DD + NR) {
        int r = tid - NR * DD;
        float acc = ws[OFF_BASE];
        for (int j = 0; j < DD; ++j) acc += ws[OFF_T3 + r * DD + j] * W_a[j];
        ws[OFF_T3A + r] = acc;
    }
}

// ---------------- edge pass: one wave32 per edge ----------------
// score_e = ent[s].w1 + ent[d].w2 + t3a[r]; b_e = exp(leaky_relu(score))
// b_sum[s] += b_e; A_dst[s] += b_e*ent[d]; A_rel[s] += b_e*T3[r]; out_rel[e] = R_out[r]
__global__ void __launch_bounds__(256)
kgat_edge(const int* __restrict__ trip, const float* __restrict__ ent,
          float* __restrict__ ws, float* __restrict__ out_rel) {
    int wid  = (int)((blockIdx.x * blockDim.x + threadIdx.x) >> 5);   // edge id
    int lane = threadIdx.x & 31;
    if (wid >= NE) return;

    int s = trip[3 * wid + 0];
    int d = trip[3 * wid + 1];
    int r = trip[3 * wid + 2];

    const float4 es  = *(const float4*)(ent + (size_t)s * DD + lane * 4);
    const float4 ed  = *(const float4*)(ent + (size_t)d * DD + lane * 4);
    const float4 vw1 = *(const float4*)(ws + OFF_W1 + lane * 4);
    const float4 vw2 = *(const float4*)(ws + OFF_W2 + lane * 4);

    float p = es.x * vw1.x + es.y * vw1.y + es.z * vw1.z + es.w * vw1.w
            + ed.x * vw2.x + ed.y * vw2.y + ed.z * vw2.z + ed.w * vw2.w;
    // wave32 xor reduction -> all lanes hold the sum
    for (int m = 16; m >= 1; m >>= 1) p += __shfl_xor(p, m, 32);

    float score = p + ws[OFF_T3A + r];
    float lrelu = score > 0.0f ? score : 0.01f * score;
    float be = __expf(lrelu);

    if (lane == 0) atomicAdd(ws + OFF_BSUM + s, be);

    const float4 t3 = *(const float4*)(ws + OFF_T3 + (size_t)r * DD + lane * 4);
    size_t so = (size_t)s * DD + lane * 4;
    atomicAdd(ws + OFF_ADST + so + 0, be * ed.x);
    atomicAdd(ws + OFF_ADST + so + 1, be * ed.y);
    atomicAdd(ws + OFF_ADST + so + 2, be * ed.z);
    atomicAdd(ws + OFF_ADST + so + 3, be * ed.w);
    atomicAdd(ws + OFF_AREL + so + 0, be * t3.x);
    atomicAdd(ws + OFF_AREL + so + 1, be * t3.y);
    atomicAdd(ws + OFF_AREL + so + 2, be * t3.z);
    atomicAdd(ws + OFF_AREL + so + 3, be * t3.w);

    const float4 ro = *(const float4*)(ws + OFF_ROUT + (size_t)r * DD + lane * 4);
    *(float4*)(out_rel + (size_t)wid * DD + lane * 4) = ro;
}

// ---------------- node pass: WMMA f32 16x16x4 ----------------
// h[n] = mask * ( ent[n]@M1 + (A_dst[n]@M2 + A_rel[n]) / b_sum[n] + bias_c )
// One wave computes a 16-row x 128-col output block; 8 column tiles of 16x16.
__global__ void __launch_bounds__(256)
kgat_node(const float* __restrict__ ent, const float* __restrict__ ws,
          float* __restrict__ out_h) {
    int wave = (int)((blockIdx.x * blockDim.x + threadIdx.x) >> 5);
    int lane = threadIdx.x & 31;
    int row0 = wave * 16;
    if (row0 >= NN) return;          // uniform per wave -> EXEC all-ones inside

    const float* M1P   = ws + OFF_M1P;
    const float* M2P   = ws + OFF_M2P;
    const float* A_dst = ws + OFF_ADST;
    const float* A_rel = ws + OFF_AREL;
    const float* b_sum = ws + OFF_BSUM;
    const float* biasc = ws + OFF_BIASC;

    int half = lane >> 4;            // 0: K={0,1}, 1: K={2,3} within a k-step
    int lrow = lane & 15;

    // hoisted per-output-row terms (row depends only on v and half, not nt)
    float inv[8], flag[8];
    #pragma unroll
    for (int v = 0; v < 8; ++v) {
        int rrow = row0 + v + 8 * half;
        float bs = b_sum[rrow];
        bool has = bs > 0.0f;
        inv[v]  = has ? 1.0f / bs : 0.0f;
        flag[v] = has ? 1.0f : 0.0f;
    }

    for (int nt = 0; nt < 8; ++nt) {
        int col = nt * 16 + lrow;
        v8f acc1 = {};               // ent @ M1
        v8f acc2 = {};               // A_dst @ M2
        for (int kt = 0; kt < 32; ++kt) {
            int k = kt * 4 + 2 * half;
            v2f a1 = *(const v2f*)(ent   + (size_t)(row0 + lrow) * DD + k);
            v2f a2 = *(const v2f*)(A_dst + (size_t)(row0 + lrow) * DD + k);
            // packed pair layout: one b64 load per B fragment
            v2f b1 = *(const v2f*)(M1P + (size_t)(kt * 2 + half) * 256 + col * 2);
            v2f b2 = *(const v2f*)(M2P + (size_t)(kt * 2 + half) * 256 + col * 2);
            acc1 = __builtin_amdgcn_wmma_f32_16x16x4_f32(false, a1, false, b1,
                                                         (short)0, acc1, false, false);
            acc2 = __builtin_amdgcn_wmma_f32_16x16x4_f32(false, a2, false, b2,
                                                         (short)0, acc2, false, false);
        }
        // epilogue: D element v of lane -> row = row0 + v + 8*half, col = nt*16 + lrow
        float bc = biasc[col];
        #pragma unroll
        for (int v = 0; v < 8; ++v) {
            int rrow = row0 + v + 8 * half;
            // branch-free: A_dst/A_rel rows are all-zero when b_sum==0
            float val = flag[v] * (acc1[v] + bc)
                      + (acc2[v] + A_rel[(size_t)rrow * DD + col]) * inv[v];
            out_h[(size_t)rrow * DD + col] = val;
        }
    }
}

extern "C" void kernel_launch(void* const* d_in, const int* in_sizes, int n_in,
                              void* d_out, int out_size, void* d_ws, size_t ws_size,
                              hipStream_t stream) {
    (void)in_sizes; (void)n_in; (void)out_size; (void)ws_size;
    const int*   trip      = (const int*)  d_in[0];
    const float* ent       = (const float*)d_in[1];
    const float* rel_embed = (const float*)d_in[2];
    const float* W_ent     = (const float*)d_in[3];
    const float* b_ent     = (const float*)d_in[4];
    const float* W_rel     = (const float*)d_in[5];
    const float* b_rel     = (const float*)d_in[6];
    const float* W_fc      = (const float*)d_in[7];
    const float* b_fc      = (const float*)d_in[8];
    const float* W_a       = (const float*)d_in[9];
    const float* b_a       = (const float*)d_in[10];
    const float* W_rel2    = (const float*)d_in[11];
    const float* b_rel2    = (const float*)d_in[12];

    float* ws      = (float*)d_ws;
    float* out_h   = (float*)d_out;
    float* out_rel = out_h + (size_t)NN * DD;

    // 1) zero accumulators (A_dst, A_rel, b_sum are contiguous at ws start)
    size_t nzero = (size_t)2 * NN * DD + NN;
    kgat_zero<<<2048, 256, 0, stream>>>(ws, nzero);

    // 2) fused weight products
    kgat_prep_M<<<(3 * DD * DD + DD + 255) / 256, 256, 0, stream>>>(
        W_ent, W_rel, W_fc, b_ent, b_rel, b_fc, ws);

    // 3) attention projection vectors
    kgat_prep_w<<<2, 256, 0, stream>>>(W_a, b_a, ws);

    // 4) relation tables
    kgat_prep_relA<<<(2 * NR * DD + 255) / 256, 256, 0, stream>>>(rel_embed, W_rel, b_rel, ws);
    kgat_prep_relB<<<(NR * DD + NR + 255) / 256, 256, 0, stream>>>(W_rel2, b_rel2, W_a, ws);

    // 5) edge pass: 8 edges per 256-thread block (wave32 per edge)
    kgat_edge<<<(NE + 7) / 8, 256, 0, stream>>>(trip, ent, ws, out_rel);

    // 6) node pass: WMMA GEMM, 16 rows per wave -> 6250 waves
    int nwaves = (NN + 15) / 16;
    kgat_node<<<(nwaves + 7) / 8, 256, 0, stream>>>(ent, ws, out_h);
}